// ImageEntropyLoss_36189394436766
// MI455X (gfx1250) — compile-verified
//
#include <hip/hip_runtime.h>
#include <hip/hip_bf16.h>

// ---------------------------------------------------------------------------
// ImageEntropyLoss forward for MI455X (gfx1250).
// Bandwidth-bound batched 256-bin histogram (100.7 MB in, scalar out,
// ~4.3 us roofline at 23.3 TB/s). Strategy:
//   * depth-4 async global->LDS pipeline (ASYNCcnt + s_wait_asynccnt)
//   * per-wave (wave32) privatized u32 histograms in LDS (ds_add_u32)
//   * integer-only atomics => deterministic result
//   * entropy is linear over (image,bin) pairs -> single flat reduction
// hist_kernel is first in the file so the disasm snippet shows the hot loop.
// ---------------------------------------------------------------------------

#ifndef __has_builtin
#define __has_builtin(x) 0
#endif

#if defined(__gfx1250__) && \
    __has_builtin(__builtin_amdgcn_global_load_async_to_lds_b128) && \
    __has_builtin(__builtin_amdgcn_s_wait_asynccnt)
#define USE_ASYNC 1
#else
#define USE_ASYNC 0
#endif

#define BINS 256
#define IMGN (512 * 512)          // pixels per image
#define THREADS 256
#define WAVES 8                   // 256 / wave32
#define STAGE_PIX 1024            // 256 threads * 4 floats
#define NSTAGES 8                 // stages per block
#define PPB (STAGE_PIX * NSTAGES) // 8192 pixels per block
#define BLOCKS_PER_IMG (IMGN / PPB) // 32
#define DEPTH 4                   // async pipeline depth (stages in flight)

#if USE_ASYNC
typedef int v4i_ __attribute__((vector_size(16)));
typedef __attribute__((address_space(1))) v4i_ *gv4i_p;
typedef __attribute__((address_space(3))) v4i_ *lv4i_p;
#define ASYNC_B128(gp, lp)                                                   \
  __builtin_amdgcn_global_load_async_to_lds_b128((gv4i_p)(gp), (lv4i_p)(lp), \
                                                 0, 0)
#define WAIT_ASYNC(n) __builtin_amdgcn_s_wait_asynccnt(n)
#endif

// ---------------------------------------------------------------------------
__global__ void __launch_bounds__(THREADS)
hist_kernel(const float *__restrict__ x, unsigned *__restrict__ g_cnt) {
  __shared__ __align__(16) float buf[DEPTH][3][STAGE_PIX]; // 48 KB
  __shared__ unsigned hist[WAVES][BINS];                   // 8 KB

  const int tid = threadIdx.x;
  const int wave = tid >> 5;
  const int lane = tid & 31;
  const int img = blockIdx.x / BLOCKS_PER_IMG;
  const int chunk = blockIdx.x % BLOCKS_PER_IMG;
  const int lane_off = tid * 4; // float offset within a stage

  // Each wave owns (and zeroes) its private histogram row: no barrier needed
  // until the final merge.
  for (int i = lane; i < BINS; i += 32) hist[wave][i] = 0u;

  // Channel base pointers for this block's pixel range.
  const float *chanR =
      x + (size_t)img * 3 * IMGN + (size_t)chunk * PPB;

  const float interval = 1.0f / 255.0f;
  const float eps = 0.5f * interval;

  unsigned *hrow = hist[wave];

  auto proc = [&](float r, float g, float b) {
    float Y = 0.257f * r + 0.564f * g + 0.098f * b + 0.0625f;
    float idxf = rintf(Y * 255.0f);           // nearest bin (round-half-even)
    float v = idxf * interval;                // bin center value
    int idx = (int)idxf;
    bool in = (Y > v - eps) & (Y < v + eps) & (idx >= 0) & (idx <= BINS - 1);
    if (in) atomicAdd(&hrow[idx], 1u);        // ds_add_u32
  };

  auto consume4 = [&](const float *p) {
    float4 r = *(const float4 *)(p + 0);
    float4 g = *(const float4 *)(p + STAGE_PIX);
    float4 b = *(const float4 *)(p + 2 * STAGE_PIX);
    proc(r.x, g.x, b.x);
    proc(r.y, g.y, b.y);
    proc(r.z, g.z, b.z);
    proc(r.w, g.w, b.w);
  };

#if USE_ASYNC
  auto issue = [&](int s) {
    const int bi = s & (DEPTH - 1);
    const float *g0 = chanR + (size_t)s * STAGE_PIX + lane_off;
    ASYNC_B128(g0, &buf[bi][0][lane_off]);
    ASYNC_B128(g0 + IMGN, &buf[bi][1][lane_off]);
    ASYNC_B128(g0 + 2 * IMGN, &buf[bi][2][lane_off]);
  };
  auto consume = [&](int s) {
    const int bi = s & (DEPTH - 1);
    consume4(&buf[bi][0][lane_off]); // ds_load_b128 x3
  };

  // Depth-4 pipeline; 3 async ops per wave per stage, loads complete in
  // order, each thread consumes only the lanes it staged -> per-wave
  // s_wait_asynccnt is sufficient (no __syncthreads in the hot loop).
  issue(0); issue(1); issue(2); issue(3);
  WAIT_ASYNC(9); consume(0); issue(4);
  WAIT_ASYNC(9); consume(1); issue(5);
  WAIT_ASYNC(9); consume(2); issue(6);
  WAIT_ASYNC(9); consume(3); issue(7);
  WAIT_ASYNC(9); consume(4);
  WAIT_ASYNC(6); consume(5);
  WAIT_ASYNC(3); consume(6);
  WAIT_ASYNC(0); consume(7);
#else
  // Fallback: direct 128-bit global loads, no LDS staging.
  for (int s = 0; s < NSTAGES; ++s) {
    const float *g0 = chanR + (size_t)s * STAGE_PIX + lane_off;
    float4 r = *(const float4 *)(g0);
    float4 g = *(const float4 *)(g0 + IMGN);
    float4 b = *(const float4 *)(g0 + 2 * IMGN);
    proc(r.x, g.x, b.x);
    proc(r.y, g.y, b.y);
    proc(r.z, g.z, b.z);
    proc(r.w, g.w, b.w);
  }
#endif

  __syncthreads();

  // Merge the 8 per-wave histograms; one u32 global atomic per bin per block.
  unsigned total = 0;
  const int bin = tid; // THREADS == BINS
#pragma unroll
  for (int w = 0; w < WAVES; ++w) total += hist[w][bin];
  if (total) atomicAdd(&g_cnt[img * BINS + bin], total);
}

// ---------------------------------------------------------------------------
__global__ void __launch_bounds__(THREADS)
zero_counts_kernel(unsigned *__restrict__ g_cnt, int n) {
  int i = blockIdx.x * blockDim.x + threadIdx.x;
  if (i < n) g_cnt[i] = 0u;
}

// ---------------------------------------------------------------------------
// entropy = sign * (1/B) * sum over all (img,bin) with cnt>=1 of -H*log2(H),
// H = cnt/N.  Linear in (img,bin) -> one flat fixed-order reduction.
__global__ void __launch_bounds__(THREADS)
finalize_kernel(const unsigned *__restrict__ g_cnt,
                const int *__restrict__ sign, float *__restrict__ out,
                int nimg) {
  const int tid = threadIdx.x;
  const float invN = 1.0f / (float)IMGN;
  float acc = 0.0f;
  const int total = nimg * BINS;
  for (int i = tid; i < total; i += THREADS) {
    unsigned c = g_cnt[i];
    if (c >= 1u) {
      float H = (float)c * invN;
      acc -= H * log2f(H);
    }
  }
  __shared__ float red[THREADS];
  red[tid] = acc;
  __syncthreads();
  for (int off = THREADS / 2; off > 0; off >>= 1) {
    if (tid < off) red[tid] += red[tid + off];
    __syncthreads();
  }
  if (tid == 0)
    out[0] = (float)sign[0] * red[0] * (1.0f / (float)nimg);
}

// ---------------------------------------------------------------------------
extern "C" void kernel_launch(void *const *d_in, const int *in_sizes, int n_in,
                              void *d_out, int out_size, void *d_ws,
                              size_t ws_size, hipStream_t stream) {
  (void)n_in; (void)out_size; (void)ws_size;
  const float *x = (const float *)d_in[0];
  const int *sign = (const int *)d_in[1];
  float *out = (float *)d_out;
  unsigned *g_cnt = (unsigned *)d_ws; // nimg*256 u32 counts

  const int nimg = in_sizes[0] / (3 * IMGN); // 32
  const int ncnt = nimg * BINS;

  zero_counts_kernel<<<(ncnt + THREADS - 1) / THREADS, THREADS, 0, stream>>>(
      g_cnt, ncnt);
  hist_kernel<<<nimg * BLOCKS_PER_IMG, THREADS, 0, stream>>>(x, g_cnt);
  finalize_kernel<<<1, THREADS, 0, stream>>>(g_cnt, sign, out, nimg);
}